// EdgeNet_50611894616130
// MI455X (gfx1250) — compile-verified
//
#include <hip/hip_runtime.h>
#include <math.h>

typedef __attribute__((ext_vector_type(2))) float v2f;
typedef __attribute__((ext_vector_type(8))) float v8f;

#define WAVES 4
#define THREADS (WAVES * 32)                      /* 128 */
#define EDGES_PER_WAVE 32                         /* two 16-row WMMA M-tiles */
#define EDGES_PER_PASS (WAVES * EDGES_PER_WAVE)   /* 128 */
#define CHUNKS 8
#define EPB (EDGES_PER_PASS * CHUNKS)             /* 1024 edges per block */

/* k-group-of-4 interleaved layouts:
   element (k, n) -> [k>>2][ n*4 + (k&3) ]
   A lane's WMMA fragment (K = 4*ks + half*2 .. +1) is 2 contiguous dwords -> ds_load_b64,
   halves hit disjoint bank sets {4l,4l+1} vs {4l+2,4l+3}. */
#define W_ROW 512    /* W1/W2 row: 128 n * 4            */
#define H_ROW 132    /* X/H row: 32 m * 4, +4 pad so (row mod 64)=4 -> conflict-free stores */

__device__ __forceinline__ v8f wmma_f32_16x16x4(v2f a, v2f b, v8f c) {
  // D = A(16x4 f32) x B(4x16 f32) + C(16x16 f32), full fp32 matrix path
  return __builtin_amdgcn_wmma_f32_16x16x4_f32(false, a, false, b, (short)0, c, false, false);
}

__device__ __forceinline__ v2f lds2(const float* p) { return *(const v2f*)p; }

// order-preserving mapping float -> uint for atomicMax
__device__ __forceinline__ unsigned fkey(float f) {
  unsigned u = __float_as_uint(f);
  return (u & 0x80000000u) ? ~u : (u | 0x80000000u);
}
__device__ __forceinline__ float funkey(unsigned k) {
  unsigned u = (k & 0x80000000u) ? (k & 0x7fffffffu) : ~k;
  return __uint_as_float(u);
}

// ---------------------------------------------------------------------------
// Edge MLP: logits[e] = w3 . relu(W2^T relu(W1^T x_e + b1) + b2) + b3
// One wave processes 32 edges (2 M-tiles sharing B fragments) per pass.
// ---------------------------------------------------------------------------
__global__ __launch_bounds__(THREADS) void edge_mlp_kernel(
    const float* __restrict__ uf, const float* __restrict__ gf,
    const float* __restrict__ ef, const int* __restrict__ uidx,
    const int* __restrict__ gidx,
    const float* __restrict__ W1, const float* __restrict__ b1,
    const float* __restrict__ W2, const float* __restrict__ b2,
    const float* __restrict__ w3, const float* __restrict__ b3,
    float* __restrict__ logits, int E)
{
  extern __shared__ float smem[];
  float* sW1g = smem;                       // 2  rows * W_ROW = 1024
  float* sb1  = sW1g + 2 * W_ROW;           // 128
  float* sW2g = sb1 + 128;                  // 32 rows * W_ROW = 16384
  float* sb2  = sW2g + 32 * W_ROW;          // 128
  float* sw3  = sb2 + 128;                  // 128
  float* sX   = sw3 + 128;                  // WAVES * 2  * H_ROW
  float* sH   = sX + WAVES * 2 * H_ROW;     // WAVES * 32 * H_ROW

  const int tid = threadIdx.x;

  // ---- stage weights into interleaved LDS layout ----
  for (int i = tid; i < 128; i += THREADS) { sb1[i] = b1[i]; sb2[i] = b2[i]; sw3[i] = w3[i]; }
  for (int i = tid; i < 8 * 128; i += THREADS) {
    const int k = i >> 7, n = i & 127;
    sW1g[(k >> 2) * W_ROW + n * 4 + (k & 3)] = W1[i];
  }
  for (int i = tid; i < 128 * 128; i += THREADS) {
    const int k = i >> 7, n = i & 127;
    sW2g[(k >> 2) * W_ROW + n * 4 + (k & 3)] = W2[i];
  }
  __syncthreads();

  const int wid  = tid >> 5;
  const int lane = tid & 31;
  const int l16  = lane & 15;
  const int half = lane >> 4;
  const int koff = half * 2;                // lane's k&3 offset within a k-group
  float* sXw = sX + wid * (2 * H_ROW);
  float* sHw = sH + wid * (32 * H_ROW);

  const int blockBase = blockIdx.x * EPB;

  for (int c = 0; c < CHUNKS; ++c) {
    const int eBase = blockBase + c * EDGES_PER_PASS + wid * EDGES_PER_WAVE;

    // ---- stage X (32 edges x 8 feats), interleaved [k>>2][m*4 + (k&3)] ----
    {
      const int m = lane;                   // one edge per lane
      int e = eBase + m; if (e >= E) e = E - 1;   // clamp tail (uniform flow)
      const int uI = uidx[e], gI = gidx[e];
      float v[8];
      v[0] = uf[uI * 3 + 0]; v[1] = uf[uI * 3 + 1]; v[2] = uf[uI * 3 + 2];
      v[3] = gf[gI * 3 + 0]; v[4] = gf[gI * 3 + 1]; v[5] = gf[gI * 3 + 2];
      v[6] = ef[e * 2 + 0];  v[7] = ef[e * 2 + 1];
      #pragma unroll
      for (int f = 0; f < 8; ++f)
        sXw[(f >> 2) * H_ROW + m * 4 + (f & 3)] = v[f];
    }
    __builtin_amdgcn_wave_barrier();

    // ---- layer 1: [32x8] @ [8x128] + b1, relu -> sHw ----
    #pragma unroll
    for (int nt = 0; nt < 8; ++nt) {
      const int n = nt * 16 + l16;
      const v2f b0 = lds2(&sW1g[0 * W_ROW + n * 4 + koff]);   // K 0..3
      const v2f b1v = lds2(&sW1g[1 * W_ROW + n * 4 + koff]);  // K 4..7
      const float bias = sb1[n];
      #pragma unroll
      for (int t = 0; t < 2; ++t) {
        const v2f a0 = lds2(&sXw[0 * H_ROW + (t * 16 + l16) * 4 + koff]);
        const v2f a1 = lds2(&sXw[1 * H_ROW + (t * 16 + l16) * 4 + koff]);
        v8f cc = {0.f, 0.f, 0.f, 0.f, 0.f, 0.f, 0.f, 0.f};
        cc = wmma_f32_16x16x4(a0, b0, cc);
        cc = wmma_f32_16x16x4(a1, b1v, cc);
        #pragma unroll
        for (int r = 0; r < 8; ++r) {       // D row r -> M=r (+8 upper half)
          float h = cc[r] + bias;
          h = h > 0.f ? h : 0.f;
          const int m = t * 16 + r + half * 8;
          sHw[(n >> 2) * H_ROW + m * 4 + (n & 3)] = h;
        }
      }
    }
    __builtin_amdgcn_wave_barrier();

    // ---- layer 2: two M-tiles share every B fragment; relu + w3 dot fused ----
    float accw0[8], accw1[8];
    #pragma unroll
    for (int r = 0; r < 8; ++r) { accw0[r] = 0.f; accw1[r] = 0.f; }

    for (int nt = 0; nt < 8; ++nt) {
      const int n = nt * 16 + l16;
      v8f c0 = {0.f, 0.f, 0.f, 0.f, 0.f, 0.f, 0.f, 0.f};
      v8f c1 = {0.f, 0.f, 0.f, 0.f, 0.f, 0.f, 0.f, 0.f};
      #pragma unroll 4
      for (int ks = 0; ks < 32; ++ks) {     // K = 4*ks .. 4*ks+3
        const v2f b  = lds2(&sW2g[ks * W_ROW + n * 4 + koff]);
        const v2f a0 = lds2(&sHw[ks * H_ROW + l16 * 4 + koff]);
        const v2f a1 = lds2(&sHw[ks * H_ROW + (16 + l16) * 4 + koff]);
        c0 = wmma_f32_16x16x4(a0, b, c0);   // independent accumulators:
        c1 = wmma_f32_16x16x4(a1, b, c1);   // no D->A hazard back-to-back
      }
      const float b2v = sb2[n];
      const float w3v = sw3[n];
      #pragma unroll
      for (int r = 0; r < 8; ++r) {
        float h0 = c0[r] + b2v; h0 = h0 > 0.f ? h0 : 0.f;
        float h1 = c1[r] + b2v; h1 = h1 > 0.f ? h1 : 0.f;
        accw0[r] += h0 * w3v;
        accw1[r] += h1 * w3v;
      }
    }

    // ---- reduce w3-dot partials over the 16 lanes of each half ----
    #pragma unroll
    for (int r = 0; r < 8; ++r) {
      float v0 = accw0[r], v1 = accw1[r];
      v0 += __shfl_xor(v0, 1);  v1 += __shfl_xor(v1, 1);
      v0 += __shfl_xor(v0, 2);  v1 += __shfl_xor(v1, 2);
      v0 += __shfl_xor(v0, 4);  v1 += __shfl_xor(v1, 4);
      v0 += __shfl_xor(v0, 8);  v1 += __shfl_xor(v1, 8);
      accw0[r] = v0; accw1[r] = v1;
    }
    if (l16 == 0) {                         // lanes 0 and 16 write 16 logits each
      const float bb3 = b3[0];
      #pragma unroll
      for (int r = 0; r < 8; ++r) {
        const int e0 = eBase + half * 8 + r;
        const int e1 = eBase + 16 + half * 8 + r;
        if (e0 < E) logits[e0] = accw0[r] + bb3;
        if (e1 < E) logits[e1] = accw1[r] + bb3;
      }
    }
    __builtin_amdgcn_wave_barrier();
  }
}

// ---------------------------------------------------------------------------
// NO_TX MLP per UAV: logit_no = wn2 . relu(Wn1^T u + bn1) + bn2 ; init m_key
// ---------------------------------------------------------------------------
__global__ void notx_kernel(const float* __restrict__ uf,
                            const float* __restrict__ Wn1, const float* __restrict__ bn1,
                            const float* __restrict__ wn2, const float* __restrict__ bn2,
                            float* __restrict__ logit_no, unsigned* __restrict__ mkey, int nU)
{
  __shared__ float sW[192], sB[64], sw2[64];
  const int tid = threadIdx.x;
  for (int i = tid; i < 192; i += blockDim.x) sW[i] = Wn1[i];
  for (int i = tid; i < 64; i += blockDim.x) { sB[i] = bn1[i]; sw2[i] = wn2[i]; }
  __syncthreads();
  const int u = blockIdx.x * blockDim.x + tid;
  if (u >= nU) return;
  const float f0 = uf[u * 3 + 0], f1 = uf[u * 3 + 1], f2 = uf[u * 3 + 2];
  float acc = bn2[0];
  #pragma unroll 8
  for (int j = 0; j < 64; ++j) {
    float h = f0 * sW[j] + f1 * sW[64 + j] + f2 * sW[128 + j] + sB[j];
    h = h > 0.f ? h : 0.f;
    acc += h * sw2[j];
  }
  logit_no[u] = acc;
  mkey[u] = fkey(acc);   // m = max(seg_max, logit_no): seed with logit_no
}

// ---------------------------------------------------------------------------
// Segment softmax over {edges of u} U {NO_TX}
// ---------------------------------------------------------------------------
__global__ void segmax_kernel(const float* __restrict__ logits, const int* __restrict__ uidx,
                              unsigned* __restrict__ mkey, int E)
{
  const int e = blockIdx.x * blockDim.x + threadIdx.x;
  if (e < E) atomicMax(&mkey[uidx[e]], fkey(logits[e]));
}

__global__ void user_exp_kernel(const float* __restrict__ logit_no, const unsigned* __restrict__ mkey,
                                float* __restrict__ denom, float* __restrict__ outNo, int nU)
{
  const int u = blockIdx.x * blockDim.x + threadIdx.x;
  if (u < nU) {
    const float m  = funkey(mkey[u]);
    const float en = expf(logit_no[u] - m);
    denom[u] = en;      // init denominator with exp_no
    outNo[u] = en;
  }
}

__global__ void edge_exp_kernel(const float* __restrict__ logits, const int* __restrict__ uidx,
                                const unsigned* __restrict__ mkey, float* __restrict__ denom,
                                float* __restrict__ outE, int E)
{
  const int e = blockIdx.x * blockDim.x + threadIdx.x;
  if (e < E) {
    const int u = uidx[e];
    const float ex = expf(logits[e] - funkey(mkey[u]));
    outE[e] = ex;
    atomicAdd(&denom[u], ex);
  }
}

__global__ void edge_div_kernel(float* __restrict__ outE, const int* __restrict__ uidx,
                                const float* __restrict__ denom, int E)
{
  const int e = blockIdx.x * blockDim.x + threadIdx.x;
  if (e < E) outE[e] = outE[e] / denom[uidx[e]];
}

__global__ void user_div_kernel(float* __restrict__ outNo, const float* __restrict__ denom, int nU)
{
  const int u = blockIdx.x * blockDim.x + threadIdx.x;
  if (u < nU) outNo[u] = outNo[u] / denom[u];
}

// ---------------------------------------------------------------------------
extern "C" void kernel_launch(void* const* d_in, const int* in_sizes, int n_in,
                              void* d_out, int out_size, void* d_ws, size_t ws_size,
                              hipStream_t stream)
{
  const float* uf  = (const float*)d_in[0];
  const float* gf  = (const float*)d_in[1];
  const float* ef  = (const float*)d_in[2];
  const int*   ui  = (const int*)d_in[3];
  const int*   gi  = (const int*)d_in[4];
  const float* W1  = (const float*)d_in[5];
  const float* b1  = (const float*)d_in[6];
  const float* W2  = (const float*)d_in[7];
  const float* b2  = (const float*)d_in[8];
  const float* w3  = (const float*)d_in[9];
  const float* b3  = (const float*)d_in[10];
  const float* Wn1 = (const float*)d_in[11];
  const float* bn1 = (const float*)d_in[12];
  const float* wn2 = (const float*)d_in[13];
  const float* bn2 = (const float*)d_in[14];

  const int nU = in_sizes[0] / 3;
  const int E  = in_sizes[3];

  // workspace: edge_logits[E] | logit_no[nU] | m_key[nU] (u32) | denom[nU]
  float* ws          = (float*)d_ws;
  float* edge_logits = ws;
  float* logit_no    = ws + (size_t)E;
  unsigned* mkey     = (unsigned*)(ws + (size_t)E + (size_t)nU);
  float* denom       = ws + (size_t)E + 2 * (size_t)nU;

  float* outE  = (float*)d_out;       // edge_probs [E]
  float* outNo = outE + (size_t)E;    // notx_probs [nU]

  // 1) NO_TX MLP + seed per-user running max
  notx_kernel<<<(nU + 255) / 256, 256, 0, stream>>>(uf, Wn1, bn1, wn2, bn2, logit_no, mkey, nU);

  // 2) edge MLP on the fp32 WMMA path; ~140 KB LDS/block (WGP has 320 KB)
  const size_t lds = (size_t)(2 * W_ROW + 128 + 32 * W_ROW + 128 + 128
                              + WAVES * 2 * H_ROW + WAVES * 32 * H_ROW) * sizeof(float);
  const int nBlk = (E + EPB - 1) / EPB;
  edge_mlp_kernel<<<nBlk, THREADS, lds, stream>>>(uf, gf, ef, ui, gi,
                                                  W1, b1, W2, b2, w3, b3, edge_logits, E);

  // 3..6) segment softmax
  segmax_kernel<<<(E + 255) / 256, 256, 0, stream>>>(edge_logits, ui, mkey, E);
  user_exp_kernel<<<(nU + 255) / 256, 256, 0, stream>>>(logit_no, mkey, denom, outNo, nU);
  edge_exp_kernel<<<(E + 255) / 256, 256, 0, stream>>>(edge_logits, ui, mkey, denom, outE, E);
  edge_div_kernel<<<(E + 255) / 256, 256, 0, stream>>>(outE, ui, denom, E);
  user_div_kernel<<<(nU + 255) / 256, 256, 0, stream>>>(outNo, denom, nU);
}